// minGRU2_5308579577996
// MI455X (gfx1250) — compile-verified
//
#include <hip/hip_runtime.h>
#include <hip/hip_bf16.h>
#include <cstdint>

// ---------------------------------------------------------------------------
// Bidirectional minGRU scan, x:[8,512,8192] f32 -> out:[8,256,8192] f32.
//
// Per (batch, channel, direction) sequence (2048 total), the log-space
// reference reduces to the linear recurrence
//     z = sigmoid(gate); a = 1-z; b = z * (h>=0 ? 1+h : exp(h));
//     y_t = a_t * y_{t-1} + b_t,  y_{-1} = 0
// (backward half scanned in reversed time).
//
// Kernel structure (memory-bound; 192 MiB traffic ~= 8.2 us @ 23.3 TB/s):
//   * 1 workgroup (1024 threads = 32 wave32) per sequence, 2048 blocks.
//   * TDM tensor_load_to_lds stages the h-row + gate-row (2 x 8192 f32,
//     row pitch 128*L elements) into 64 KB LDS with one descriptor,
//     waited via s_wait_tensorcnt. Fallback: direct float4 global loads.
//   * Each thread serially folds its 8-element chunk into an affine pair
//     (A,B); wave32 shuffle Kogge-Stone scan + 32-entry LDS scan composes
//     pairs block-wide; threads replay their chunk with the carry.
// ---------------------------------------------------------------------------

namespace {
constexpr int Bx  = 8;
constexpr int Hx  = 512;
constexpr int Cx  = 128;          // channels per (direction, role)
constexpr int Lx  = 8192;
constexpr int TPB = 1024;
constexpr int EPT = Lx / TPB;     // 8 elements per thread
}

#if defined(__HIP_DEVICE_COMPILE__) && \
    __has_builtin(__builtin_amdgcn_tensor_load_to_lds) && \
    __has_builtin(__builtin_amdgcn_s_wait_tensorcnt)
#define MINGRU_USE_TDM 1
#else
#define MINGRU_USE_TDM 0
#endif

typedef __attribute__((ext_vector_type(4))) unsigned int tdm_v4u;
typedef __attribute__((ext_vector_type(8))) int          tdm_v8i;
typedef __attribute__((ext_vector_type(4))) int          tdm_v4i;

__global__ __launch_bounds__(TPB)
void mingru_bidir_scan(const float* __restrict__ x, float* __restrict__ out)
{
  const int blk = blockIdx.x;         // 0..2047
  const int dir = blk >> 10;          // 0 = forward half, 1 = backward half
  const int r   = blk & 1023;
  const int b   = r >> 7;             // / Cx
  const int c   = r & (Cx - 1);

  const float* __restrict__ hrow =
      x + ((size_t)b * Hx + (size_t)dir * 2 * Cx + (size_t)c) * (size_t)Lx;
  const float* __restrict__ grow = hrow + (size_t)Cx * (size_t)Lx;
  float* __restrict__ orow =
      out + ((size_t)b * 2 * Cx + (size_t)dir * Cx + (size_t)c) * (size_t)Lx;

  const int tid  = (int)threadIdx.x;
  const int lane = tid & 31;
  const int wid  = tid >> 5;

  __shared__ float sA[32];
  __shared__ float sB[32];

#if MINGRU_USE_TDM
  __shared__ __align__(16) float stage[2 * Lx];   // 64 KB: [0,L)=h, [L,2L)=gate
  if (wid == 0) {
    // ---- Tensor DMA descriptor (D#), 2D tile: 8192 x 2 rows of f32 ----
    const unsigned long long ga = (unsigned long long)(uintptr_t)hrow;
    const unsigned ldsaddr = (unsigned)(uintptr_t)(&stage[0]); // low 32b of flat LDS addr = LDS byte offset
    tdm_v4u g0;
    g0[0] = 1u;                                               // count=1, user mode
    g0[1] = ldsaddr;                                          // lds_addr
    g0[2] = (unsigned)(ga & 0xffffffffull);                   // global_addr[31:0]
    g0[3] = (unsigned)((ga >> 32) & 0x01ffffffull) | (2u << 30); // global_addr[56:32] | type=2
    tdm_v8i g1;
    g1[0] = (int)(2u << 16);                                  // data_size = 4 bytes
    g1[1] = (int)((unsigned)Lx << 16);                        // tensor_dim0[15:0] @ bits 63:48
    g1[2] = (int)(((unsigned)Lx >> 16) | (2u << 16));         // tensor_dim0[31:16] | tensor_dim1=2
    g1[3] = (int)((unsigned)Lx << 16);                        // tile_dim0 @ bits 127:112
    g1[4] = 2;                                                // tile_dim1 = 2 (h row + gate row)
    g1[5] = (int)(unsigned)(Cx * Lx);                         // tensor_dim0_stride[31:0] = 1048576 elems
    g1[6] = 0;                                                // stride hi16 | dim1_stride lo16
    g1[7] = 0;
    tdm_v4i gz4 = {0, 0, 0, 0};                               // groups 2/3 unused (2D)
    tdm_v8i gz8 = {0, 0, 0, 0, 0, 0, 0, 0};                   // trailing group (6-arg form)
    __builtin_amdgcn_tensor_load_to_lds(g0, g1, gz4, gz4, gz8, 0);
    __builtin_amdgcn_s_wait_tensorcnt(0);
  }
  __syncthreads();
#endif

  const int j0    = tid * EPT;                   // logical (scan-order) base
  const int pbase = dir ? (Lx - j0 - EPT) : j0;  // physical base in memory

#if MINGRU_USE_TDM
  const float* hsrc = &stage[pbase];
  const float* gsrc = &stage[Lx + pbase];
#else
  const float* hsrc = hrow + pbase;
  const float* gsrc = grow + pbase;
#endif
  float4 h0 = *(const float4*)(hsrc);
  float4 h1 = *(const float4*)(hsrc + 4);
  float4 q0 = *(const float4*)(gsrc);
  float4 q1 = *(const float4*)(gsrc + 4);
  float hv[EPT] = {h0.x, h0.y, h0.z, h0.w, h1.x, h1.y, h1.z, h1.w};
  float gv[EPT] = {q0.x, q0.y, q0.z, q0.w, q1.x, q1.y, q1.z, q1.w};

  // Per-element (a,b) in logical scan order (reversed for dir==1).
  float av[EPT], bv[EPT];
#pragma unroll
  for (int k = 0; k < EPT; ++k) {
    const int m = dir ? (EPT - 1 - k) : k;
    const float h = hv[m], g = gv[m];
    const float e   = __expf(-g);                        // e^{-gate}
    const float inv = __builtin_amdgcn_rcpf(1.0f + e);   // sigmoid(gate)
    const float a   = e * inv;                           // 1 - z
    const float gg  = (h >= 0.0f) ? (h + 1.0f) : __expf(h);
    av[k] = a;
    bv[k] = inv * gg;                                    // z * g(h)
  }

  // Thread-local affine fold: y_out = A*y_in + B over the 8-element chunk.
  float A = 1.0f, Bv = 0.0f;
#pragma unroll
  for (int k = 0; k < EPT; ++k) { Bv = fmaf(av[k], Bv, bv[k]); A *= av[k]; }

  // Wave32 inclusive Kogge-Stone scan of (A,B); composition:
  // prev (Ap,Bp) then cur (A,B) -> (Ap*A, A*Bp + B).
  float Ai = A, Bi = Bv;
#pragma unroll
  for (int d = 1; d < 32; d <<= 1) {
    const float Ap = __shfl_up(Ai, d, 32);
    const float Bp = __shfl_up(Bi, d, 32);
    if (lane >= d) { Bi = fmaf(Ai, Bp, Bi); Ai *= Ap; }
  }
  if (lane == 31) { sA[wid] = Ai; sB[wid] = Bi; }
  __syncthreads();

  // Wave 0 scans the 32 per-wave aggregates, stores EXCLUSIVE prefixes.
  if (wid == 0) {
    float Aw = sA[lane], Bw = sB[lane];
#pragma unroll
    for (int d = 1; d < 32; d <<= 1) {
      const float Ap = __shfl_up(Aw, d, 32);
      const float Bp = __shfl_up(Bw, d, 32);
      if (lane >= d) { Bw = fmaf(Aw, Bp, Bw); Aw *= Ap; }
    }
    float Axe = __shfl_up(Aw, 1, 32);
    float Bxe = __shfl_up(Bw, 1, 32);
    if (lane == 0) { Axe = 1.0f; Bxe = 0.0f; }
    sA[lane] = Axe; sB[lane] = Bxe;
  }
  __syncthreads();

  // Carry into this thread: waves-before  then  lanes-before (y_start = 0).
  const float Bw = sB[wid];
  float Al = __shfl_up(Ai, 1, 32);
  float Bl = __shfl_up(Bi, 1, 32);
  if (lane == 0) { Al = 1.0f; Bl = 0.0f; }
  float y = fmaf(Al, Bw, Bl);

  // Replay chunk with carry; scatter back to physical order.
  float o[EPT];
#pragma unroll
  for (int k = 0; k < EPT; ++k) {
    y = fmaf(av[k], y, bv[k]);
    const int m = dir ? (EPT - 1 - k) : k;
    o[m] = y;
  }
  float4* dp = (float4*)(orow + pbase);
  dp[0] = make_float4(o[0], o[1], o[2], o[3]);
  dp[1] = make_float4(o[4], o[5], o[6], o[7]);
}

extern "C" void kernel_launch(void* const* d_in, const int* in_sizes, int n_in,
                              void* d_out, int out_size, void* d_ws, size_t ws_size,
                              hipStream_t stream) {
  (void)in_sizes; (void)n_in; (void)out_size; (void)d_ws; (void)ws_size;
  const float* x = (const float*)d_in[0];
  float* out = (float*)d_out;
  dim3 grid(2 * Bx * Cx);   // 2048 sequences
  dim3 block(TPB);          // 32 wave32 per sequence
  hipLaunchKernelGGL(mingru_bidir_scan, grid, block, 0, stream, x, out);
}